// MultiVectorEncoder_52664888983642
// MI455X (gfx1250) — compile-verified
//
#include <hip/hip_runtime.h>
#include <hip/hip_bf16.h>
#include <math.h>

#define G_GRID    37
#define NPATCH    (G_GRID * G_GRID)        // 1369
#define NCUE      5
#define S_TOK     (NCUE + NPATCH)          // 1374
#define D_DIM     768
#define NTILE     ((NPATCH + 15) / 16)     // 86
#define WAVES_PB  8
#define BLOCKS_PB ((NTILE + WAVES_PB - 1) / WAVES_PB)  // 11
#define LDS_STRIDE 772                     // 768 + 4 pad -> conflict-free A loads
#define NTOK_OUT  (2 * NCUE)               // 10

typedef float v2f __attribute__((ext_vector_type(2)));
typedef float v8f __attribute__((ext_vector_type(8)));

// ---------------- kernel 0: init packed argmax workspace --------------------
__global__ void init_ws_kernel(unsigned long long* __restrict__ ws, int n) {
    int i = blockIdx.x * blockDim.x + threadIdx.x;
    if (i < n) ws[i] = 0ull;   // 0 is below any packed (key<<32 | ~idx)
}

// ---------------- kernel 1: sim GEMM via WMMA f32 16x16x4 + argmax ----------
__global__ __launch_bounds__(256) void sim_argmax_kernel(
        const float* __restrict__ tokens,
        unsigned long long* __restrict__ ws) {
    __shared__ float cue_lds[16 * LDS_STRIDE];   // rows 5..15 zero (A padding)

    const int b   = blockIdx.x / BLOCKS_PB;
    const int blk = blockIdx.x % BLOCKS_PB;
    const float* base = tokens + (size_t)b * S_TOK * D_DIM;

    // zero the padded A tile, then fill rows 0..4 with the cues
    for (int i = threadIdx.x; i < 16 * LDS_STRIDE; i += 256) cue_lds[i] = 0.0f;
    __syncthreads();
    for (int i = threadIdx.x; i < NCUE * D_DIM; i += 256) {
        int c = i / D_DIM, d = i % D_DIM;
        cue_lds[c * LDS_STRIDE + d] = base[c * D_DIM + d];
    }
    __syncthreads();

    const int wave = threadIdx.x >> 5;
    const int lane = threadIdx.x & 31;
    const int tile = blk * WAVES_PB + wave;
    if (tile >= NTILE) return;               // whole-wave uniform exit

    const int nl   = lane & 15;              // M row (A) / N col (B)
    const int koff = (lane < 16) ? 0 : 2;    // K sub-offset per lane half

    int n_global = tile * 16 + nl;
    int n_clamp  = min(n_global, NPATCH - 1);
    const float* pcol = base + (size_t)(NCUE + n_clamp) * D_DIM; // patch row
    const float* arow = cue_lds + nl * LDS_STRIDE;

    v8f acc = {};
    #pragma unroll 4
    for (int k0 = 0; k0 < D_DIM; k0 += 4) {
        v2f a, bm;
        a.x  = arow[k0 + koff];
        a.y  = arow[k0 + koff + 1];
        bm.x = pcol[k0 + koff];
        bm.y = pcol[k0 + koff + 1];
        acc = __builtin_amdgcn_wmma_f32_16x16x4_f32(
                  false, a, false, bm, (short)0, acc, false, false);
    }

    // lanes 0..15: acc[c] = sim[c][tile*16 + lane] for c = 0..7 (we need 0..4)
    bool valid = (lane < 16) && (n_global < NPATCH);
    #pragma unroll
    for (int c = 0; c < NCUE; ++c) {
        float    v = valid ? acc[c] : -INFINITY;
        unsigned i = (unsigned)n_global;
        #pragma unroll
        for (int off = 8; off >= 1; off >>= 1) {   // reduce within 16-lane half
            float    ov = __shfl_xor(v, off, 32);
            unsigned oi = __shfl_xor(i, off, 32);
            if (ov > v || (ov == v && oi < i)) { v = ov; i = oi; }
        }
        if (lane == 0) {
            unsigned fb  = __float_as_uint(v);
            unsigned key = (fb & 0x80000000u) ? ~fb : (fb | 0x80000000u);
            unsigned long long packed =
                ((unsigned long long)key << 32) | (unsigned long long)(~i);
            atomicMax(&ws[b * NCUE + c], packed);  // max key, tie -> min idx
        }
    }
}

// ---------------- kernel 2: 3x3 ROI mean + L2 normalize ---------------------
__global__ __launch_bounds__(256) void pool_norm_kernel(
        const float* __restrict__ tokens,
        const unsigned long long* __restrict__ ws,
        float* __restrict__ out) {
    __shared__ float red[256];
    const int b = blockIdx.x / NTOK_OUT;
    const int t = blockIdx.x % NTOK_OUT;
    const float* base = tokens + (size_t)b * S_TOK * D_DIM;

    float vals[3];
    if (t < NCUE) {
        const float* src = base + (size_t)t * D_DIM;
        #pragma unroll
        for (int j = 0; j < 3; ++j) vals[j] = src[threadIdx.x + j * 256];
    } else {
        const int c = t - NCUE;
        unsigned long long p = ws[b * NCUE + c];
        int idx = (int)(~(unsigned)(p & 0xFFFFFFFFull));
        int h = idx / G_GRID, w = idx % G_GRID;
        int h0 = max(h - 1, 0), h1 = min(h + 1, G_GRID - 1);
        int w0 = max(w - 1, 0), w1 = min(w + 1, G_GRID - 1);
        float inv_cnt = 1.0f / (float)((h1 - h0 + 1) * (w1 - w0 + 1));
        #pragma unroll
        for (int j = 0; j < 3; ++j) vals[j] = 0.0f;
        for (int hh = h0; hh <= h1; ++hh)
            for (int ww = w0; ww <= w1; ++ww) {
                const float* p2 = base + (size_t)(NCUE + hh * G_GRID + ww) * D_DIM;
                #pragma unroll
                for (int j = 0; j < 3; ++j) vals[j] += p2[threadIdx.x + j * 256];
            }
        #pragma unroll
        for (int j = 0; j < 3; ++j) vals[j] *= inv_cnt;
    }

    red[threadIdx.x] = vals[0]*vals[0] + vals[1]*vals[1] + vals[2]*vals[2];
    __syncthreads();
    for (int s = 128; s > 0; s >>= 1) {
        if (threadIdx.x < s) red[threadIdx.x] += red[threadIdx.x + s];
        __syncthreads();
    }
    float scale = 1.0f / fmaxf(sqrtf(red[0]), 1e-12f);

    float* dst = out + ((size_t)b * NTOK_OUT + t) * D_DIM;
    #pragma unroll
    for (int j = 0; j < 3; ++j)
        dst[threadIdx.x + j * 256] = vals[j] * scale;
}

// ---------------- host launch ----------------------------------------------
extern "C" void kernel_launch(void* const* d_in, const int* in_sizes, int n_in,
                              void* d_out, int out_size, void* d_ws, size_t ws_size,
                              hipStream_t stream) {
    const float* tokens = (const float*)d_in[0];
    float* out = (float*)d_out;
    unsigned long long* ws = (unsigned long long*)d_ws;

    const int B = in_sizes[0] / (S_TOK * D_DIM);   // 128
    const int nslots = B * NCUE;

    init_ws_kernel<<<(nslots + 255) / 256, 256, 0, stream>>>(ws, nslots);
    sim_argmax_kernel<<<B * BLOCKS_PB, 256, 0, stream>>>(tokens, ws);
    pool_norm_kernel<<<B * NTOK_OUT, 256, 0, stream>>>(tokens, ws, out);
}